// Net_87780541596381
// MI455X (gfx1250) — compile-verified
//
#include <hip/hip_runtime.h>

typedef __attribute__((ext_vector_type(2))) float v2f;
typedef __attribute__((ext_vector_type(8))) float v8f;

#define NEG_SLOPE 0.01f

// ---------------------------------------------------------------- utilities
__global__ __launch_bounds__(256) void zero_kernel(float* __restrict__ p, int n) {
    int i = blockIdx.x * blockDim.x + threadIdx.x;
    if (i < n) p[i] = 0.0f;
}

__global__ __launch_bounds__(256) void degree_kernel(const int* __restrict__ dst,
                                                     float* __restrict__ deg, int n_edges) {
    int e = blockIdx.x * blockDim.x + threadIdx.x;
    if (e < n_edges) atomicAdd(deg + dst[e], 1.0f);
}

// ------------------------------------------------- layer 1 node transform (fin=2)
// Y[n, 0:16]=x@A1, [16:32]=x@B1, [32:48]=x@C1, [48:64]=x@root1
__global__ __launch_bounds__(256) void node_transform_l1(
    const float* __restrict__ x,      // [N,2]
    const float* __restrict__ nnw,    // [32,2]
    const float* __restrict__ nnb,    // [32]
    const float* __restrict__ root,   // [2,16]
    float* __restrict__ Y, int n_nodes)
{
    int gid = blockIdx.x * blockDim.x + threadIdx.x;
    int n = gid >> 6;
    int o = gid & 63;
    if (n >= n_nodes) return;
    int blk = o >> 4, oo = o & 15;
    float m0, m1;
    if (blk == 0)      { m0 = nnw[(oo) * 2 + 0];      m1 = nnw[(16 + oo) * 2 + 0]; }
    else if (blk == 1) { m0 = nnw[(oo) * 2 + 1];      m1 = nnw[(16 + oo) * 2 + 1]; }
    else if (blk == 2) { m0 = nnb[oo];                m1 = nnb[16 + oo];           }
    else               { m0 = root[oo];               m1 = root[16 + oo];          }
    float x0 = x[n * 2 + 0], x1 = x[n * 2 + 1];
    Y[(size_t)n * 64 + o] = fmaf(x0, m0, x1 * m1);
}

// ------------------------------------------- layers 2-4 node transform (WMMA, fin=16)
// One wave per 16-node tile: D[16x64] = Xtile[16x16] @ [A|B|C|root]  (4 x 16x16 blocks)
// Uses V_WMMA_F32_16X16X4_F32, accumulating 4 K-steps of 4.
__global__ __launch_bounds__(256) void node_transform16_wmma(
    const float* __restrict__ Xin,    // rows at Xin[n*64 + k], k=0..15 (column block of X5)
    const float* __restrict__ nnw,    // [256,2]
    const float* __restrict__ nnb,    // [256]
    const float* __restrict__ root,   // [16,16]
    float* __restrict__ Y, int n_nodes)
{
    int wave  = (blockIdx.x * blockDim.x + threadIdx.x) >> 5;
    int lane  = threadIdx.x & 31;
    int node0 = wave * 16;
    if (node0 >= n_nodes) return;          // wave-uniform

    int m     = lane & 15;                 // A: row; B/D: column
    int khalf = (lane < 16) ? 0 : 2;       // K-offset for upper half-wave
    const float* xrow = Xin + (size_t)(node0 + m) * 64;

    v8f acc0 = {}, acc1 = {}, acc2 = {}, acc3 = {};
    #pragma unroll
    for (int s = 0; s < 4; ++s) {
        int kb = s * 4 + khalf;
        v2f a; a.x = xrow[kb]; a.y = xrow[kb + 1];
        int i0 = (kb * 16 + m), i1 = ((kb + 1) * 16 + m);
        v2f b0, b1, b2, b3;
        b0.x = nnw[i0 * 2 + 0]; b0.y = nnw[i1 * 2 + 0];
        b1.x = nnw[i0 * 2 + 1]; b1.y = nnw[i1 * 2 + 1];
        b2.x = nnb[i0];         b2.y = nnb[i1];
        b3.x = root[i0];        b3.y = root[i1];
        acc0 = __builtin_amdgcn_wmma_f32_16x16x4_f32(false, a, false, b0, (short)0, acc0, false, false);
        acc1 = __builtin_amdgcn_wmma_f32_16x16x4_f32(false, a, false, b1, (short)0, acc1, false, false);
        acc2 = __builtin_amdgcn_wmma_f32_16x16x4_f32(false, a, false, b2, (short)0, acc2, false, false);
        acc3 = __builtin_amdgcn_wmma_f32_16x16x4_f32(false, a, false, b3, (short)0, acc3, false, false);
    }
    int rbase = (lane < 16) ? 0 : 8;
    #pragma unroll
    for (int j = 0; j < 8; ++j) {
        float* yr = Y + (size_t)(node0 + rbase + j) * 64;
        yr[ 0 + m] = acc0[j];
        yr[16 + m] = acc1[j];
        yr[32 + m] = acc2[j];
        yr[48 + m] = acc3[j];
    }
}

// --------------------------------------------- layer 5 node transform (WMMA, fin=64)
// Y5[n,0:8] = x5 @ [A3|B3|C3|root3]  (64x8, padded to 64x16, staged in LDS)
__global__ __launch_bounds__(256) void node_transform64_wmma(
    const float* __restrict__ X5,     // [N,64]
    const float* __restrict__ nnw,    // [128,2]
    const float* __restrict__ nnb,    // [128]
    const float* __restrict__ root,   // [64,2]
    float* __restrict__ Y5, int n_nodes)
{
    __shared__ float M5[64 * 16];
    for (int i = threadIdx.x; i < 64 * 16; i += blockDim.x) {
        int k = i >> 4, col = i & 15;
        float v = 0.0f;
        if (col < 2)      v = nnw[(k * 2 + col) * 2 + 0];
        else if (col < 4) v = nnw[(k * 2 + (col - 2)) * 2 + 1];
        else if (col < 6) v = nnb[k * 2 + (col - 4)];
        else if (col < 8) v = root[k * 2 + (col - 6)];
        M5[i] = v;
    }
    __syncthreads();

    int wave  = (blockIdx.x * blockDim.x + threadIdx.x) >> 5;
    int lane  = threadIdx.x & 31;
    int node0 = wave * 16;
    if (node0 >= n_nodes) return;          // wave-uniform

    int m     = lane & 15;
    int khalf = (lane < 16) ? 0 : 2;
    const float* xrow = X5 + (size_t)(node0 + m) * 64;

    v8f acc = {};
    #pragma unroll
    for (int s = 0; s < 16; ++s) {
        int kb = s * 4 + khalf;
        v2f a; a.x = xrow[kb]; a.y = xrow[kb + 1];
        v2f b; b.x = M5[kb * 16 + m]; b.y = M5[(kb + 1) * 16 + m];
        acc = __builtin_amdgcn_wmma_f32_16x16x4_f32(false, a, false, b, (short)0, acc, false, false);
    }
    int rbase = (lane < 16) ? 0 : 8;
    if (m < 8) {
        #pragma unroll
        for (int j = 0; j < 8; ++j)
            Y5[(size_t)(node0 + rbase + j) * 8 + m] = acc[j];
    }
}

// -------------------------------------------------------- edge scatter (fout=16)
__global__ __launch_bounds__(256) void edge_scatter16(
    const int* __restrict__ src, const int* __restrict__ dst,
    const float* __restrict__ ea, const float* __restrict__ Y,
    float* __restrict__ agg, int n_edges)
{
    int gid = blockIdx.x * blockDim.x + threadIdx.x;
    int e = gid >> 4, o = gid & 15;
    if (e >= n_edges) return;
    int s = src[e], d = dst[e];
    float a0 = ea[2 * e], a1 = ea[2 * e + 1];
    const float* ys = Y + (size_t)s * 64;
    float v = fmaf(a0, ys[o], fmaf(a1, ys[16 + o], ys[32 + o]));
    atomicAdd(agg + (size_t)d * 16 + o, v);
}

// --------------------------------------------------------- edge scatter (fout=2)
__global__ __launch_bounds__(256) void edge_scatter2(
    const int* __restrict__ src, const int* __restrict__ dst,
    const float* __restrict__ ea, const float* __restrict__ Y5,
    float* __restrict__ agg2, int n_edges)
{
    int e = blockIdx.x * blockDim.x + threadIdx.x;
    if (e >= n_edges) return;
    int s = src[e], d = dst[e];
    float a0 = ea[2 * e], a1 = ea[2 * e + 1];
    const float* y = Y5 + (size_t)s * 8;
    atomicAdd(agg2 + d * 2 + 0, fmaf(a0, y[0], fmaf(a1, y[2], y[4])));
    atomicAdd(agg2 + d * 2 + 1, fmaf(a0, y[1], fmaf(a1, y[3], y[5])));
}

// ---------------------------------------------------- finalize: mean + root + bias
__global__ __launch_bounds__(256) void finalize16(
    const float* __restrict__ agg, const float* __restrict__ deg,
    const float* __restrict__ Y, const float* __restrict__ bias,
    float* __restrict__ Xout /* column block of X5, stride 64 */, int n_nodes)
{
    int gid = blockIdx.x * blockDim.x + threadIdx.x;
    int n = gid >> 4, o = gid & 15;
    if (n >= n_nodes) return;
    float c = fmaxf(deg[n], 1.0f);
    float v = agg[(size_t)n * 16 + o] / c + Y[(size_t)n * 64 + 48 + o] + bias[o];
    v = (v > 0.0f) ? v : v * NEG_SLOPE;      // LeakyReLU
    Xout[(size_t)n * 64 + o] = v;
}

__global__ __launch_bounds__(256) void finalize2(
    const float* __restrict__ agg2, const float* __restrict__ deg,
    const float* __restrict__ Y5, const float* __restrict__ bias,
    float* __restrict__ out, int n_nodes)
{
    int gid = blockIdx.x * blockDim.x + threadIdx.x;
    int n = gid >> 1, o = gid & 1;
    if (n >= n_nodes) return;
    float c = fmaxf(deg[n], 1.0f);
    out[n * 2 + o] = agg2[n * 2 + o] / c + Y5[(size_t)n * 8 + 6 + o] + bias[o];
}

// ---------------------------------------------------------------------- driver
extern "C" void kernel_launch(void* const* d_in, const int* in_sizes, int n_in,
                              void* d_out, int out_size, void* d_ws, size_t ws_size,
                              hipStream_t stream) {
    const float* x     = (const float*)d_in[0];
    const int*   eidx  = (const int*)  d_in[1];
    const float* ea    = (const float*)d_in[2];
    const float* nn1w  = (const float*)d_in[3];
    const float* nn1b  = (const float*)d_in[4];
    const float* root1 = (const float*)d_in[5];
    const float* bias1 = (const float*)d_in[6];
    const float* nn2w  = (const float*)d_in[7];
    const float* nn2b  = (const float*)d_in[8];
    const float* root2 = (const float*)d_in[9];
    const float* bias2 = (const float*)d_in[10];
    const float* nn3w  = (const float*)d_in[11];
    const float* nn3b  = (const float*)d_in[12];
    const float* root3 = (const float*)d_in[13];
    const float* bias3 = (const float*)d_in[14];
    float* out = (float*)d_out;

    const int N = in_sizes[0] / 2;
    const int E = in_sizes[1] / 2;
    const int* src = eidx;
    const int* dst = eidx + E;

    float* ws  = (float*)d_ws;
    float* X5  = ws;                            // [N,64] concat(x1..x4)
    float* Y   = X5 + (size_t)N * 64;           // [N,64] per-layer node transforms
    float* agg = Y  + (size_t)N * 64;           // [N,16]
    float* deg = agg + (size_t)N * 16;          // [N]

    const int B = 256;
    int gridZeroAgg = (N * 16 + B - 1) / B;
    int gridZeroDeg = (N + B - 1) / B;
    int gridEdges   = (E + B - 1) / B;
    int gridE16     = (E * 16 + B - 1) / B;
    int gridN64     = (N * 64 + B - 1) / B;
    int gridN16     = (N * 16 + B - 1) / B;
    int gridN2      = (N * 2 + B - 1) / B;
    int tiles       = (N + 15) / 16;
    int gridWmma    = (tiles + (B / 32) - 1) / (B / 32);

    // degrees (graph fixed across all layers)
    zero_kernel<<<gridZeroDeg, B, 0, stream>>>(deg, N);
    degree_kernel<<<gridEdges, B, 0, stream>>>(dst, deg, E);

    // ---- layer 1 (fin=2 -> 16)
    zero_kernel<<<gridZeroAgg, B, 0, stream>>>(agg, N * 16);
    node_transform_l1<<<gridN64, B, 0, stream>>>(x, nn1w, nn1b, root1, Y, N);
    edge_scatter16<<<gridE16, B, 0, stream>>>(src, dst, ea, Y, agg, E);
    finalize16<<<gridN16, B, 0, stream>>>(agg, deg, Y, bias1, X5 + 0, N);

    // ---- layers 2..4 (fin=16 -> 16, shared nn2/root2)
    for (int l = 0; l < 3; ++l) {
        zero_kernel<<<gridZeroAgg, B, 0, stream>>>(agg, N * 16);
        node_transform16_wmma<<<gridWmma, B, 0, stream>>>(X5 + l * 16, nn2w, nn2b, root2, Y, N);
        edge_scatter16<<<gridE16, B, 0, stream>>>(src, dst, ea, Y, agg, E);
        finalize16<<<gridN16, B, 0, stream>>>(agg, deg, Y, bias2, X5 + (l + 1) * 16, N);
    }

    // ---- layer 5 (fin=64 -> 2)
    zero_kernel<<<gridN2, B, 0, stream>>>(agg, N * 2);
    node_transform64_wmma<<<gridWmma, B, 0, stream>>>(X5, nn3w, nn3b, root3, Y, N);
    edge_scatter2<<<gridEdges, B, 0, stream>>>(src, dst, ea, Y, agg, E);
    finalize2<<<gridN2, B, 0, stream>>>(agg, deg, Y, bias3, out, N);
}